// DistanceMatrix_43593918055020
// MI455X (gfx1250) — compile-verified
//
#include <hip/hip_runtime.h>
#include <cstdint>

#define BATCH 8
#define H 256
#define W 256
#define INF_VAL 1.0e10f
#define THR 0.5f
#define SCALE 50.0f

// ---- CDNA5 async global->LDS helpers (inline asm: portable across both toolchains) ----

// Low 32 bits of a generic pointer to LDS == wave-relative LDS byte offset
// (per CDNA5 aperture mapping: the LDS address is the low 32 bits of the address).
__device__ __forceinline__ uint32_t lds_off_of(const void* p) {
    return (uint32_t)(uintptr_t)p;
}

// GLOBAL_LOAD_ASYNC_TO_LDS_B128: per-lane 16B  memory -> LDS, tracked by ASYNCcnt.
__device__ __forceinline__ void async_load_b128(uint32_t lds_off, const void* gaddr) {
    asm volatile("global_load_async_to_lds_b128 %0, %1, off"
                 :
                 : "v"(lds_off), "v"((uint64_t)(uintptr_t)gaddr)
                 : "memory");
}

__device__ __forceinline__ void wait_asynccnt0() {
    asm volatile("s_wait_asynccnt 0" ::: "memory");
}

// ---------------------------------------------------------------------------
// Pass 1 (columns): tmp[b,y,xi] = min_xq (xi-xq)^2 + m[b,y,xq]
// One block = 4 rows. Thread t: row r = t>>6, 4 outputs xi = (t&63) + 64k
// sharing every LDS broadcast read of m[r][q]  (1 ds_load : 8 VALU).
// ---------------------------------------------------------------------------
__global__ __launch_bounds__(256) void edt_pass1(const float* __restrict__ mask,
                                                 float* __restrict__ tmp) {
    __shared__ float s_m[4][W];                       // 4 KB
    const int tid  = threadIdx.x;
    const int row0 = blockIdx.x * 4;                  // flattened b*H + y, 4 rows/block

    // Stage 4 rows (4 KB) into LDS: 256 lanes x one b128 (16 B) each.
    {
        const int r_ld   = tid >> 6;                  // 0..3
        const int chunk  = tid & 63;                  // 64 x 16B chunks per row
        const float* g   = mask + (size_t)(row0 + r_ld) * W + chunk * 4;
        async_load_b128(lds_off_of(&s_m[r_ld][chunk * 4]), g);
        wait_asynccnt0();
    }
    __syncthreads();

    // Threshold in place (each thread owns 4 flat slots).
    {
        float* flat = &s_m[0][0];
        #pragma unroll
        for (int i = 0; i < 4; ++i) {
            int idx = tid + 256 * i;
            flat[idx] = (flat[idx] > THR) ? 0.0f : INF_VAL;
        }
    }
    __syncthreads();

    const int r  = tid >> 6;
    const int xl = tid & 63;
    float best[4];
    #pragma unroll
    for (int k = 0; k < 4; ++k) best[k] = 3.0e38f;

    #pragma unroll 2
    for (int q = 0; q < W; ++q) {
        const float m  = s_m[r][q];                   // wave-wide broadcast read
        const float fq = (float)q;
        #pragma unroll
        for (int k = 0; k < 4; ++k) {
            float dx = (float)(xl + 64 * k) - fq;
            best[k] = fminf(best[k], __fmaf_rn(dx, dx, m));
        }
    }

    float* out_row = tmp + (size_t)(row0 + r) * W;
    #pragma unroll
    for (int k = 0; k < 4; ++k) out_row[xl + 64 * k] = best[k];
}

// ---------------------------------------------------------------------------
// Pass 2 (rows): d[b,yi,x] = sqrt( min_yq (yi-yq)^2 + tmp[b,yq,x] )
// One block = 32-column strip x 64 output rows of one image; full 256x32
// strip (32 KB) staged in LDS. Thread: col c = t&31, 8 yi outputs sharing
// every LDS read (conflict-free: 32 consecutive banks per wave).
// Also reduces the global max of d (wave32 shfl + int atomicMax, valid >=0).
// ---------------------------------------------------------------------------
__global__ __launch_bounds__(256) void edt_pass2(const float* __restrict__ tmp,
                                                 float* __restrict__ dist,
                                                 int* __restrict__ gmax) {
    __shared__ float s_col[H * 32];                   // 32 KB
    const int yt  = blockIdx.x;                       // 0..3   (64 rows each)
    const int xt  = blockIdx.y;                       // 0..7   (32 cols each)
    const int b   = blockIdx.z;                       // 0..7
    const int tid = threadIdx.x;
    const int x0  = xt * 32;
    const int y0  = yt * 64;

    // Stage tmp[b, 0:256, x0:x0+32] -> LDS via async b128 (8 per thread).
    {
        const int chunk = tid & 7;                    // 8 x 16B chunks per row
        const int ybase = tid >> 3;                   // 32 rows per sweep
        const float* base = tmp + (size_t)b * H * W + x0;
        #pragma unroll
        for (int i = 0; i < 8; ++i) {
            int y = ybase + 32 * i;
            async_load_b128(lds_off_of(&s_col[y * 32 + chunk * 4]),
                            base + (size_t)y * W + chunk * 4);
        }
        wait_asynccnt0();
    }
    __syncthreads();

    const int c    = tid & 31;
    const int ysub = tid >> 5;                        // 0..7
    float d2[8];
    #pragma unroll
    for (int j = 0; j < 8; ++j) d2[j] = 3.0e38f;

    #pragma unroll 2
    for (int yq = 0; yq < H; ++yq) {
        const float m   = s_col[yq * 32 + c];         // 1 ds_load : 16 VALU
        const float fyq = (float)yq;
        #pragma unroll
        for (int j = 0; j < 8; ++j) {
            float dy = (float)(y0 + ysub + 8 * j) - fyq;
            d2[j] = fminf(d2[j], __fmaf_rn(dy, dy, m));
        }
    }

    float lmax = 0.0f;
    #pragma unroll
    for (int j = 0; j < 8; ++j) {
        const int yi = y0 + ysub + 8 * j;
        const float d = sqrtf(d2[j]);
        dist[((size_t)b * H + yi) * W + x0 + c] = d;  // coalesced 128B per wave
        lmax = fmaxf(lmax, d);
    }

    // wave32 max reduction, then one atomic per wave.
    #pragma unroll
    for (int off = 16; off > 0; off >>= 1)
        lmax = fmaxf(lmax, __shfl_xor(lmax, off, 32));
    if ((tid & 31) == 0)
        atomicMax(gmax, __float_as_int(lmax));        // d >= 0 -> int order == float order
}

// ---------------------------------------------------------------------------
// Finalize: out = 2 * sigmoid( -d / (max+1) * 50 ), in place on d_out.
// ---------------------------------------------------------------------------
__global__ __launch_bounds__(256) void edt_finalize(float* __restrict__ out,
                                                    const int* __restrict__ gmax,
                                                    int n) {
    const int i = blockIdx.x * blockDim.x + threadIdx.x;
    if (i >= n) return;
    const float maxd = __int_as_float(*gmax) + 1.0f;
    const float z = out[i] / maxd * SCALE;
    out[i] = 2.0f / (1.0f + expf(z));                 // sigmoid(-z)*2
}

extern "C" void kernel_launch(void* const* d_in, const int* in_sizes, int n_in,
                              void* d_out, int out_size, void* d_ws, size_t ws_size,
                              hipStream_t stream) {
    (void)in_sizes; (void)n_in; (void)ws_size;
    const float* mask = (const float*)d_in[0];        // [8,256,256] fp32
    float* out = (float*)d_out;                       // [8,256,256] fp32
    float* tmp = (float*)d_ws;                        // 2 MB scratch
    int*   gmax = (int*)((char*)d_ws + (size_t)BATCH * H * W * sizeof(float));

    const int n = BATCH * H * W;

    hipMemsetAsync(gmax, 0, sizeof(int), stream);     // deterministic per call
    edt_pass1<<<BATCH * H / 4, 256, 0, stream>>>(mask, tmp);
    dim3 g2(H / 64, W / 32, BATCH);                   // (4, 8, 8)
    edt_pass2<<<g2, 256, 0, stream>>>(tmp, out, gmax);
    edt_finalize<<<(n + 255) / 256, 256, 0, stream>>>(out, gmax, n);
}